// MCMCDA_79328045957694
// MI455X (gfx1250) — compile-verified
//
#include <hip/hip_runtime.h>
#include <hip/hip_bf16.h>
#include <stdint.h>

// ---------------------------------------------------------------------------
// MCMC with delayed acceptance, restructured for MI455X (gfx1250).
//
// Key algebraic optimization: the Gaussian likelihood is quadratic, so
//   lp_out(t) = -0.5 t^T Gp_out t + cp_out^T t + const,
//     Gp_out = I + inv_var * A^T A,  cp_out = inv_var * A^T y
// (same for the "inner" model with M = A + 0.05*sin(A)). Constants cancel in
// all MH log-ratios. This turns each 512x16 matvec into a 16x16 quadratic
// form that lives in the registers of a single wave32.
//
// Precompute: A^T A and M^T M are 16x16x512 GEMMs -> V_WMMA_F32_16X16X4_F32
// (CDNA5 full-fp32 WMMA), 128 accumulating steps each, two independent
// accumulator chains. For the symmetric Gram product the 16x4 A-operand and
// 4x16 B-operand lane layouts coincide, so one float2 load feeds both.
//
// Chain: one wave32. Lanes 0-15 carry Gp_out rows, lanes 16-31 carry Gp_in
// rows, so stage 2 evaluates BOTH log-posteriors in a single pass using
// wave32 shuffles for the broadcast-matvec and xor-tree reductions.
// ---------------------------------------------------------------------------

typedef float v2f __attribute__((ext_vector_type(2)));
typedef float v8f __attribute__((ext_vector_type(8)));

#define INV_VAR   4.0f      // 1 / NOISE^2, NOISE = 0.5
#define ITER_MCMC 100000
#define ITER_DA   5000
#define N_DA      20000
#define DT0       0.1f

// ws layout (floats): [0..255] Gp_out, [256..511] Gp_in,
//                     [512..527] cp_out, [528..543] cp_in

// ------------------------------ precompute ---------------------------------
__global__ void MCMCDA_precompute_kernel(const float* __restrict__ A,
                                         const float* __restrict__ y,
                                         float* __restrict__ ws) {
  const int lane = threadIdx.x;    // single wave32, EXEC all ones for WMMA
  const int col  = lane & 15;
  const int khi  = lane >> 4;      // 0: K-pair {0,1}, 1: K-pair {2,3}

  v8f c_out = {};
  v8f c_in  = {};
  // G = A^T A as 16x16x512 GEMM in 128 steps of V_WMMA_F32_16X16X4_F32.
  // A-operand (16x4 MxK): lane L, vgpr j <-> (M = L&15, K = 2*(L>>4)+j)
  //   = A[4s + 2*(L>>4) + j][L&15] -- identical indexing to the 4x16
  // B-operand (K = j + 2*(L>>4), N = L&15), so one load feeds both.
  for (int s = 0; s < 128; ++s) {
    const int r0 = 4 * s + 2 * khi;
    const float a0 = A[(r0 + 0) * 16 + col];
    const float a1 = A[(r0 + 1) * 16 + col];
    v2f a = {a0, a1};
    v2f mm = {a0 + 0.05f * sinf(a0), a1 + 0.05f * sinf(a1)};
    c_out = __builtin_amdgcn_wmma_f32_16x16x4_f32(
        false, a,  false, a,  (short)0, c_out, false, false);
    c_in  = __builtin_amdgcn_wmma_f32_16x16x4_f32(
        false, mm, false, mm, (short)0, c_in,  false, false);
  }

  // C/D layout: vgpr v, lanes 0-15 -> row v, lanes 16-31 -> row v+8.
  #pragma unroll
  for (int v = 0; v < 8; ++v) {
    const int row  = v + 8 * khi;
    const float id = (row == col) ? 1.0f : 0.0f;
    ws[row * 16 + col]       = id + INV_VAR * c_out[v];
    ws[256 + row * 16 + col] = id + INV_VAR * c_in[v];
  }

  // cp = inv_var * (A^T y) (lanes 0-15) / (M^T y) (lanes 16-31); runs once.
  float acc = 0.0f;
  for (int r = 0; r < 512; ++r) {
    float a = A[r * 16 + col];
    float e = khi ? (a + 0.05f * sinf(a)) : a;
    acc = fmaf(e, y[r], acc);
  }
  ws[512 + khi * 16 + col] = INV_VAR * acc;
}

// ------------------------------ RNG helpers --------------------------------
__device__ __forceinline__ uint32_t mix32(uint32_t x) {
  x ^= x >> 16; x *= 0x7FEB352Du;
  x ^= x >> 15; x *= 0x846CA68Bu;
  x ^= x >> 16;
  return x;
}
__device__ __forceinline__ uint32_t rng3(uint32_t a, uint32_t b, uint32_t c) {
  return mix32(a * 0x9E3779B9u ^ mix32(b * 0x85EBCA6Bu ^ mix32(c + 0x2545F491u)));
}
__device__ __forceinline__ float u01(uint32_t x) {        // (0,1)
  return ((float)(x >> 8) + 0.5f) * (1.0f / 16777216.0f);
}
__device__ __forceinline__ float normal01(uint32_t s1, uint32_t s2) {
  const float u1 = u01(s1), u2 = u01(s2);
  return sqrtf(-2.0f * __logf(u1)) * __cosf(6.2831853071795864f * u2);
}

// -------------------------------- chain ------------------------------------
__global__ void MCMCDA_chain_kernel(const float* __restrict__ theta0,
                                    const float* __restrict__ ws,
                                    float* __restrict__ out) {
  const int lane = threadIdx.x;    // single wave32
  const int m    = lane & 15;
  const int hi   = lane >> 4;      // 0: outer model, 1: inner model

  float* __restrict__ samples = out;
  float* __restrict__ accl    = out + (size_t)ITER_MCMC * 16;
  for (int k = lane; k < ITER_DA; k += 32) accl[k] = 0.0f;   // zero-init

  // Per-lane row of the folded precision matrix + linear term.
  float G[16];
  {
    const float* Grow = ws + hi * 256 + m * 16;
    #pragma unroll
    for (int j = 0; j < 16; ++j) G[j] = Grow[j];
  }
  const float cp = ws[512 + hi * 16 + m];

  // lp_pair: lanes 0-15 compute lp_out(t), lanes 16-31 compute lp_in(t),
  // both broadcast wave-wide. t mirrored: lane L holds t[L&15].
  auto lp_pair = [&](float t_m, float& lpo, float& lpi) {
    float v = 0.0f;
    #pragma unroll
    for (int j = 0; j < 16; ++j)
      v = fmaf(G[j], __shfl(t_m, j, 32), v);     // broadcast-matvec
    float p = t_m * (cp - 0.5f * v);
    #pragma unroll
    for (int off = 1; off < 16; off <<= 1)       // xor-tree reduce per half
      p += __shfl_xor(p, off, 32);
    lpo = __shfl(p, 0, 32);
    lpi = __shfl(p, 16, 32);
  };

  float theta = theta0[m];
  float dt = DT0;
  float lpo_cur, lpi_cur;
  lp_pair(theta, lpo_cur, lpi_cur);

  // ---- Stage 1: adaptive random-walk MH (serial, latency-bound) ----
  for (int i = 0; i < ITER_MCMC; ++i) {
    const uint32_t s = (uint32_t)i;
    const float z    = normal01(rng3(s, (uint32_t)m, 1u), rng3(s, (uint32_t)m, 2u));
    const float prop = fmaf(dt, z, theta);
    float lpo_p, lpi_p;
    lp_pair(prop, lpo_p, lpi_p);
    const float log_a = fminf(lpo_p - lpo_cur, 0.0f);
    const float a     = __expf(log_a);
    const float u     = u01(rng3(s, 0xACCE55u, 3u));
    if (u < a) { theta = prop; lpo_cur = lpo_p; lpi_cur = lpi_p; }
    dt = fmaf(dt, (a - 0.234f) / ((float)i + 1.0f), dt);   // Robbins-Monro
    if (hi == 0) samples[(size_t)i * 16 + m] = theta;      // post-update theta
  }

  // ---- Stage 2: delayed acceptance (dt frozen; both lp's per pass) ----
  int inner_mh = 0;
  for (int i = 0; i < N_DA; ++i) {
    const uint32_t s = 0x40000000u + (uint32_t)i;
    const float z    = normal01(rng3(s, (uint32_t)m, 1u), rng3(s, (uint32_t)m, 2u));
    const float prop = fmaf(dt, z, theta);
    float lpo_p, lpi_p;
    lp_pair(prop, lpo_p, lpi_p);
    const float log_a1   = fminf(lpo_p - lpo_cur, 0.0f);
    const bool pass_outer = u01(rng3(s, 0xACCE55u, 3u)) < __expf(log_a1);
    const float log_a2   = fminf(lpi_p - lpi_cur - log_a1, 0.0f);
    const bool pass_inner = u01(rng3(s, 0xACCE55u, 4u)) < __expf(log_a2);
    const bool active = inner_mh < ITER_DA;
    const bool accept = pass_outer && pass_inner && active;
    if (accept) { theta = prop; lpo_cur = lpo_p; lpi_cur = lpi_p; }
    if (pass_outer && active) {
      ++inner_mh;
      if (lane == 0) accl[inner_mh - 1] = accept ? 1.0f : 0.0f;
    }
  }
}

// ------------------------------- launcher ----------------------------------
extern "C" void kernel_launch(void* const* d_in, const int* in_sizes, int n_in,
                              void* d_out, int out_size, void* d_ws, size_t ws_size,
                              hipStream_t stream) {
  const float* A      = (const float*)d_in[0];   // 512x16
  const float* y      = (const float*)d_in[1];   // 512
  const float* theta0 = (const float*)d_in[2];   // 16
  float* ws  = (float*)d_ws;                     // 544 floats used
  float* out = (float*)d_out;                    // 100000*16 + 5000 floats

  MCMCDA_precompute_kernel<<<1, 32, 0, stream>>>(A, y, ws);
  MCMCDA_chain_kernel<<<1, 32, 0, stream>>>(theta0, ws, out);
}